// GATLayer_88527865905438
// MI455X (gfx1250) — compile-verified
//
#include <hip/hip_runtime.h>
#include <math.h>

#define IN_CH   128
#define HEADS   4
#define OUT_CH  32
#define HC      128          // HEADS*OUT_CH
#define NEG_SLOPE 0.2f

typedef float v2f __attribute__((ext_vector_type(2)));
typedef float v8f __attribute__((ext_vector_type(8)));

// ---------------------------------------------------------------- init
__global__ void k_init(float* __restrict__ out, float* __restrict__ m,
                       float* __restrict__ s, int n) {
    int t = blockIdx.x * blockDim.x + threadIdx.x;
    int tot = n * HC;
    if (t < tot) out[t] = 0.0f;
    if (t < n * HEADS) { m[t] = -INFINITY; s[t] = 0.0f; }
}

// ---------------------------------------------------------------- GEMM h = x @ W  (fp32 WMMA)
// One wave computes a 16-row x 128-col slab of h. 32 K-steps x 8 N-tiles of
// v_wmma_f32_16x16x4_f32. Exact fp32 — matches reference numerics.
__global__ void __launch_bounds__(256)
k_gemm(const float* __restrict__ x, const float* __restrict__ W,
       float* __restrict__ h, int n, int nwaves) {
    int wave = (blockIdx.x * blockDim.x + threadIdx.x) >> 5;
    if (wave >= nwaves) return;                 // wave-uniform exit, EXEC stays all-1s
    int lane = threadIdx.x & 31;
    int half = lane >> 4;                       // 0 | 1
    int l16  = lane & 15;
    int m0   = wave * 16;
    int arow = m0 + l16; if (arow > n - 1) arow = n - 1;   // clamp (per-lane, no branch)

    v8f acc[8];
    #pragma unroll
    for (int t = 0; t < 8; ++t) acc[t] = (v8f)0.0f;

    for (int k = 0; k < IN_CH; k += 4) {
        // A 16x4: VGPR v holds K = k + 2*half + v for row M = l16
        v2f a = *(const v2f*)&x[(size_t)arow * IN_CH + k + 2 * half];
        #pragma unroll
        for (int nt = 0; nt < 8; ++nt) {
            // B 4x16: VGPR v holds K = k + 2*half + v, col N = nt*16 + l16
            v2f b;
            b.x = W[(size_t)(k + 2 * half + 0) * HC + nt * 16 + l16];
            b.y = W[(size_t)(k + 2 * half + 1) * HC + nt * 16 + l16];
            acc[nt] = __builtin_amdgcn_wmma_f32_16x16x4_f32(
                false, a, false, b, (short)0, acc[nt], false, false);
        }
    }
    // C/D 16x16: VGPR v -> row M = v + 8*half, col = nt*16 + l16
    #pragma unroll
    for (int nt = 0; nt < 8; ++nt) {
        #pragma unroll
        for (int v = 0; v < 8; ++v) {
            int row = m0 + v + 8 * half;
            if (row < n) h[(size_t)row * HC + nt * 16 + l16] = acc[nt][v];
        }
    }
}

// ---------------------------------------------------------------- per-(node,head) attention dots
__global__ void k_attn(const float* __restrict__ h, const float* __restrict__ att_src,
                       const float* __restrict__ att_dst, float* __restrict__ asrc,
                       float* __restrict__ adst, int n) {
    int t = blockIdx.x * blockDim.x + threadIdx.x;
    if (t >= n * HEADS) return;
    int node = t >> 2, hd = t & 3;
    const float* hp = h + (size_t)node * HC + hd * OUT_CH;
    float s1 = 0.0f, s2 = 0.0f;
    #pragma unroll 8
    for (int c = 0; c < OUT_CH; ++c) {
        float v = hp[c];
        s1 += v * att_src[hd * OUT_CH + c];
        s2 += v * att_dst[hd * OUT_CH + c];
    }
    asrc[t] = s1;
    adst[t] = s2;
}

__device__ __forceinline__ float lrelu(float v) {
    return v >= 0.0f ? v : NEG_SLOPE * v;
}

// float atomic max via ordered-int trick (works for mixed signs, init = -inf)
__device__ __forceinline__ void atomicMaxF(float* addr, float val) {
    if (val >= 0.0f) atomicMax((int*)addr, __float_as_int(val));
    else             atomicMin((unsigned int*)addr, (unsigned int)__float_as_int(val));
}

__device__ __forceinline__ void edge_nodes(const int* __restrict__ ei, int e, int E,
                                           int& src, int& dst) {
    if (e < E) { src = ei[e]; dst = ei[E + e]; }
    else       { src = dst = e - E; }          // self loops
}

// ---------------------------------------------------------------- segment max
__global__ void k_edge_max(const int* __restrict__ ei, const float* __restrict__ asrc,
                           const float* __restrict__ adst, float* __restrict__ m,
                           int n, int E) {
    int t = blockIdx.x * blockDim.x + threadIdx.x;
    if (t >= (E + n) * HEADS) return;
    int e = t >> 2, hd = t & 3;
    int src, dst; edge_nodes(ei, e, E, src, dst);
    float v = lrelu(asrc[src * HEADS + hd] + adst[dst * HEADS + hd]);
    atomicMaxF(&m[dst * HEADS + hd], v);
}

// ---------------------------------------------------------------- segment sum of exp
__global__ void k_edge_sum(const int* __restrict__ ei, const float* __restrict__ asrc,
                           const float* __restrict__ adst, const float* __restrict__ m,
                           float* __restrict__ s, int n, int E) {
    int t = blockIdx.x * blockDim.x + threadIdx.x;
    if (t >= (E + n) * HEADS) return;
    int e = t >> 2, hd = t & 3;
    int src, dst; edge_nodes(ei, e, E, src, dst);
    float v = lrelu(asrc[src * HEADS + hd] + adst[dst * HEADS + hd]);
    float p = expf(v - m[dst * HEADS + hd]);
    atomicAdd(&s[dst * HEADS + hd], p);
}

// ---------------------------------------------------------------- weighted aggregation
// 128 threads per edge (2 edges / 256-thread block); alpha recomputed (a/m/s are
// L2/L1-resident, cheaper than spilling per-edge p). out accumulated with fp32
// global atomics (L2-resident, 192MB L2 >> 25.6MB out).
__global__ void __launch_bounds__(256)
k_aggregate(const int* __restrict__ ei, const float* __restrict__ h,
            const float* __restrict__ asrc, const float* __restrict__ adst,
            const float* __restrict__ m, const float* __restrict__ s,
            float* __restrict__ out, int n, int E) {
    int e = blockIdx.x * 2 + (threadIdx.x >> 7);
    if (e >= E + n) return;
    int c  = threadIdx.x & 127;
    int hd = c >> 5;
    int src, dst; edge_nodes(ei, e, E, src, dst);
    float v = lrelu(asrc[src * HEADS + hd] + adst[dst * HEADS + hd]);
    float alpha = expf(v - m[dst * HEADS + hd]) / s[dst * HEADS + hd];
    atomicAdd(&out[(size_t)dst * HC + c], alpha * h[(size_t)src * HC + c]);
}

// ---------------------------------------------------------------- bias + ELU epilogue
__global__ void k_finalize(float* __restrict__ out, const float* __restrict__ bias, int n) {
    int t = blockIdx.x * blockDim.x + threadIdx.x;
    if (t >= n * HC) return;
    float v = out[t] + bias[t & (HC - 1)];
    out[t] = v > 0.0f ? v : expm1f(v);
}

extern "C" void kernel_launch(void* const* d_in, const int* in_sizes, int n_in,
                              void* d_out, int out_size, void* d_ws, size_t ws_size,
                              hipStream_t stream) {
    const float* x       = (const float*)d_in[0];
    const int*   ei      = (const int*)  d_in[1];
    const float* W       = (const float*)d_in[2];
    const float* att_src = (const float*)d_in[3];
    const float* att_dst = (const float*)d_in[4];
    const float* bias    = (const float*)d_in[5];
    float* out = (float*)d_out;

    int n = in_sizes[0] / IN_CH;     // 50000
    int E = in_sizes[1] / 2;         // 1600000

    // workspace layout
    float* h    = (float*)d_ws;                  // n*128
    float* asrc = h    + (size_t)n * HC;         // n*4
    float* adst = asrc + (size_t)n * HEADS;      // n*4
    float* mbuf = adst + (size_t)n * HEADS;      // n*4
    float* sbuf = mbuf + (size_t)n * HEADS;      // n*4

    int totNC = n * HC;
    k_init<<<(totNC + 255) / 256, 256, 0, stream>>>(out, mbuf, sbuf, n);

    int nwaves = (n + 15) / 16;
    k_gemm<<<(nwaves + 7) / 8, 256, 0, stream>>>(x, W, h, n, nwaves);

    k_attn<<<(n * HEADS + 255) / 256, 256, 0, stream>>>(h, att_src, att_dst, asrc, adst, n);

    int totE = (E + n) * HEADS;
    k_edge_max<<<(totE + 255) / 256, 256, 0, stream>>>(ei, asrc, adst, mbuf, n, E);
    k_edge_sum<<<(totE + 255) / 256, 256, 0, stream>>>(ei, asrc, adst, mbuf, sbuf, n, E);

    int nedges = E + n;
    k_aggregate<<<(nedges + 1) / 2, 256, 0, stream>>>(ei, h, asrc, adst, mbuf, sbuf, out, n, E);

    k_finalize<<<(totNC + 255) / 256, 256, 0, stream>>>(out, bias, n);
}